// PtModule_76166950027772
// MI455X (gfx1250) — compile-verified
//
#include <hip/hip_runtime.h>

// y[i] = x[i]*x[i] - 1.0f   (8192x16384 fp32, pure HBM stream, ~46us roofline)
//
// CDNA5 path: 4-stage x 2-wide pipelined GLOBAL_LOAD_ASYNC_TO_LDS_B128
// (ASYNCcnt, th:TH_LOAD_NT), consume from LDS, non-temporal b128 stores.
// Hot loop: constant-immediate s_wait_asynccnt, no divergence, no division.

typedef float v4f __attribute__((ext_vector_type(4)));

#define TPB   256
#define WIDE  2                 // b128 copies per lane per stage
#define DEPTH 4                 // pipeline stages

#define TILE_V4   (WIDE * TPB)            // float4s per tile (per block)
#define BUF_BYTES (WIDE * TPB * 16)       // LDS bytes per stage

__device__ __forceinline__ void issue_pair(unsigned lds0, unsigned boff,
                                           const float* base) {
    asm volatile("global_load_async_to_lds_b128 %0, %1, %2 th:TH_LOAD_NT"
                 :: "v"(lds0), "v"(boff), "s"(base) : "memory");
    asm volatile("global_load_async_to_lds_b128 %0, %1, %2 th:TH_LOAD_NT"
                 :: "v"(lds0 + (unsigned)(TPB * 16)), "v"(boff + (unsigned)(TPB * 16)),
                    "s"(base) : "memory");
}

__device__ __forceinline__ v4f sqm1(v4f v) {
    v4f r;
    r.x = __builtin_fmaf(v.x, v.x, -1.0f);
    r.y = __builtin_fmaf(v.y, v.y, -1.0f);
    r.z = __builtin_fmaf(v.z, v.z, -1.0f);
    r.w = __builtin_fmaf(v.w, v.w, -1.0f);
    return r;
}

__global__ __launch_bounds__(TPB) void sqm1_kernel(const float* __restrict__ xf,
                                                   float* __restrict__ yf,
                                                   unsigned nvec,      // # float4
                                                   unsigned ntotal,    // # float
                                                   unsigned iterFloor, // tiles/blocks
                                                   unsigned iterRem)   // tiles%blocks
{
    __shared__ v4f tile[DEPTH][WIDE][TPB];

    const unsigned tid     = threadIdx.x;
    const unsigned gstride = gridDim.x;
    const unsigned ntiles  = nvec / TILE_V4;        // full tiles
    v4f* __restrict__ yv   = (v4f*)yf;

    // lane's LDS byte offset of stage-0/slot-0; other slots are affine
    const unsigned ldsBase = (unsigned)(size_t)&tile[0][0][tid];

    const unsigned myTiles = iterFloor + (blockIdx.x < iterRem ? 1u : 0u);

    const unsigned step = gstride * TILE_V4;        // float4 step per tile iter
    unsigned idx0     = blockIdx.x * TILE_V4 + tid; // consume index (float4)
    unsigned issueIdx = idx0;                       // issue index (float4)

    // ---- prologue: fill pipeline (up to DEPTH stages, 2 copies each) ----
    #pragma unroll
    for (unsigned k = 0; k < DEPTH; ++k) {
        if (k < myTiles)
            issue_pair(ldsBase + k * (unsigned)BUF_BYTES, issueIdx * 16u, xf);
        issueIdx += step;
    }

    // ---- steady state: full pipe -> constant wait immediate ----
    const unsigned steady = myTiles > (DEPTH - 1u) ? myTiles - (DEPTH - 1u) : 0u;
    unsigned i = 0;
    for (; i < steady; ++i) {
        asm volatile("s_wait_asynccnt 0x6" ::: "memory");   // 8 in flight -> oldest 2 done

        const unsigned buf  = i & (DEPTH - 1u);
        const unsigned lds0 = ldsBase + buf * (unsigned)BUF_BYTES;

        v4f a = tile[buf][0][tid];                  // ds_load_b128
        v4f b = tile[buf][1][tid];
        __builtin_nontemporal_store(sqm1(a), yv + idx0);
        __builtin_nontemporal_store(sqm1(b), yv + idx0 + TPB);

        if (i + DEPTH < myTiles)                    // uniform scalar branch
            issue_pair(lds0, issueIdx * 16u, xf);
        issueIdx += step;
        idx0     += step;
    }

    // ---- drain epilogue: <= DEPTH-1 tiles left, fixed wait immediates ----
    for (; i < myTiles; ++i) {
        const unsigned r = myTiles - i;             // 1..DEPTH-1 tiles outstanding
        if (r >= 3)      asm volatile("s_wait_asynccnt 0x4" ::: "memory");
        else if (r == 2) asm volatile("s_wait_asynccnt 0x2" ::: "memory");
        else             asm volatile("s_wait_asynccnt 0x0" ::: "memory");

        const unsigned buf = i & (DEPTH - 1u);
        v4f a = tile[buf][0][tid];
        v4f b = tile[buf][1][tid];
        __builtin_nontemporal_store(sqm1(a), yv + idx0);
        __builtin_nontemporal_store(sqm1(b), yv + idx0 + TPB);
        idx0 += step;
    }

    // ---- vector tail: float4s beyond the last full tile ----
    for (unsigned idx = ntiles * TILE_V4 + blockIdx.x * TPB + tid; idx < nvec;
         idx += gstride * TPB) {
        v4f v = ((const v4f*)xf)[idx];
        __builtin_nontemporal_store(sqm1(v), yv + idx);
    }

    // ---- scalar tail: floats beyond the last float4 ----
    for (unsigned j = nvec * 4u + blockIdx.x * TPB + tid; j < ntotal;
         j += gstride * TPB) {
        float v = xf[j];
        yf[j] = __builtin_fmaf(v, v, -1.0f);
    }
}

extern "C" void kernel_launch(void* const* d_in, const int* in_sizes, int n_in,
                              void* d_out, int out_size, void* d_ws, size_t ws_size,
                              hipStream_t stream) {
    const float* x = (const float*)d_in[0];
    float*       y = (float*)d_out;

    const unsigned n     = (unsigned)out_size;   // 134,217,728 floats
    const unsigned nvec  = n / 4u;               // 33,554,432 float4s
    const unsigned tiles = nvec / TILE_V4;       // 65,536 full tiles

    unsigned blocks = tiles < 4096u ? (tiles ? tiles : 1u) : 4096u;
    const unsigned iterFloor = tiles / blocks;   // 16 for this shape
    const unsigned iterRem   = tiles % blocks;   // 0  for this shape

    sqm1_kernel<<<dim3(blocks), dim3(TPB), 0, stream>>>(x, y, nvec, n,
                                                        iterFloor, iterRem);
}